// GNN_43911745634881
// MI455X (gfx1250) — compile-verified
//
#include <hip/hip_runtime.h>
#include <math.h>

typedef float v2f __attribute__((ext_vector_type(2)));
typedef float v8f __attribute__((ext_vector_type(8)));

#define KDIM 128   // inner dimension of every GEMM in this net

__device__ __forceinline__ float gelu_exact(float v) {
    return 0.5f * v * (1.0f + erff(v * 0.70710678118654752440f));
}

// ---------------- row-normalize input features (one wave per node) --------
__global__ __launch_bounds__(256)
void k_rownorm(const float* __restrict__ x, float* __restrict__ h, int n) {
    int w    = (blockIdx.x * blockDim.x + threadIdx.x) >> 5;
    int lane = threadIdx.x & 31;
    if (w >= n) return;
    const float4 v = ((const float4*)(x + (size_t)w * KDIM))[lane];
    float ss = v.x * v.x + v.y * v.y + v.z * v.z + v.w * v.w;
#pragma unroll
    for (int off = 16; off > 0; off >>= 1) ss += __shfl_xor(ss, off, 32);
    float nrm = sqrtf(ss);
    nrm = (nrm == 0.0f) ? 1e-8f : nrm;
    float s = 1.0f / nrm;
    float4 o;
    o.x = v.x * s; o.y = v.y * s; o.z = v.z * s; o.w = v.w * s;
    ((float4*)(h + (size_t)w * KDIM))[lane] = o;
}

// ---------------- degree / normalization ---------------------------------
__global__ __launch_bounds__(256)
void k_deg(const int* __restrict__ dst, float* __restrict__ deg, int E) {
    int e = blockIdx.x * blockDim.x + threadIdx.x;
    if (e < E) atomicAdd(&deg[dst[e]], 1.0f);
}

__global__ __launch_bounds__(256)
void k_dis(float* __restrict__ d, int n) {
    int i = blockIdx.x * blockDim.x + threadIdx.x;
    if (i < n) d[i] = rsqrtf(d[i] + 1.0f);   // +1 self loop; always > 0
}

// ---------------- fp32 WMMA GEMM: C[M x OC] = A[M x 128] @ W[128 x OC] ----
// Block = 128 threads (4 waves); each wave owns a 16-row tile, sweeps OC/16
// column tiles.  W is staged transposed in LDS so each lane's B fragment is
// one contiguous 8-byte ds_load.
template <int OC, bool EPILOGUE>
__global__ __launch_bounds__(128)
void k_gemm_wmma(const float* __restrict__ A, const float* __restrict__ W,
                 const float* __restrict__ bias, float* __restrict__ C, int M) {
    __shared__ float Wt[OC * KDIM];        // Wt[n][k] = W[k][n]
    for (int idx = threadIdx.x; idx < OC * KDIM; idx += blockDim.x) {
        int n = idx >> 7;          // idx / 128
        int k = idx & 127;
        Wt[n * KDIM + k] = W[k * OC + n];
    }
    __syncthreads();

    const int wave = threadIdx.x >> 5;
    const int lane = threadIdx.x & 31;
    const int m    = lane & 15;
    const int koff = (lane >> 4) << 1;     // 0 for lanes 0-15, 2 for 16-31
    const int row0 = (blockIdx.x * 4 + wave) * 16;
    if (row0 >= M) return;

    // Preload this wave's A panel as 32 fragments (ISA 16x4 f32 A layout).
    v2f a[32];
    const float* Arow = A + (size_t)(row0 + m) * KDIM + koff;
#pragma unroll
    for (int kk = 0; kk < 32; ++kk)
        a[kk] = *(const v2f*)(Arow + kk * 4);

    for (int nt = 0; nt < OC / 16; ++nt) {
        v8f acc = {0.f, 0.f, 0.f, 0.f, 0.f, 0.f, 0.f, 0.f};
        const float* wrow = &Wt[(nt * 16 + m) * KDIM + koff];
#pragma unroll
        for (int kk = 0; kk < 32; ++kk) {
            v2f b = *(const v2f*)(wrow + kk * 4);
            acc = __builtin_amdgcn_wmma_f32_16x16x4_f32(
                false, a[kk], false, b, (short)0, acc, false, false);
        }
        const int col = nt * 16 + m;
        const int rbase = row0 + ((lane >> 4) << 3);   // +8 for upper lanes
#pragma unroll
        for (int v = 0; v < 8; ++v) {
            float val = acc[v];
            if (EPILOGUE) val = gelu_exact(val + bias[col]);
            C[(size_t)(rbase + v) * OC + col] = val;
        }
    }
}

// ---------------- edge scatter-add (one wave per edge; L2-resident) -------
__global__ __launch_bounds__(256)
void k_agg_edges(const float* __restrict__ t, const int* __restrict__ src,
                 const int* __restrict__ dst, const float* __restrict__ dis,
                 float* __restrict__ acc, int E) {
    int e    = (blockIdx.x * blockDim.x + threadIdx.x) >> 5;
    int lane = threadIdx.x & 31;
    if (e >= E) return;
    int s = src[e], d = dst[e];
    float w = dis[s] * dis[d];
    const float4 v = ((const float4*)(t + (size_t)s * KDIM))[lane];
    float* out = acc + (size_t)d * KDIM + lane * 4;
    atomicAdd(out + 0, v.x * w);
    atomicAdd(out + 1, v.y * w);
    atomicAdd(out + 2, v.z * w);
    atomicAdd(out + 3, v.w * w);
}

// ------- finalize conv: acc + self-loop + bias, GELU (one wave per node) --
__global__ __launch_bounds__(256)
void k_finalize(const float* __restrict__ acc, const float* __restrict__ t,
                const float* __restrict__ dis, const float* __restrict__ bias,
                float* __restrict__ out, int n) {
    int i    = (blockIdx.x * blockDim.x + threadIdx.x) >> 5;
    int lane = threadIdx.x & 31;
    if (i >= n) return;
    float sl = dis[i]; sl *= sl;                       // self-loop weight
    const float4 a = ((const float4*)(acc + (size_t)i * KDIM))[lane];
    const float4 tv = ((const float4*)(t + (size_t)i * KDIM))[lane];
    const float4 b = ((const float4*)bias)[lane];
    float4 o;
    o.x = gelu_exact(a.x + tv.x * sl + b.x);
    o.y = gelu_exact(a.y + tv.y * sl + b.y);
    o.z = gelu_exact(a.z + tv.z * sl + b.z);
    o.w = gelu_exact(a.w + tv.w * sl + b.w);
    ((float4*)(out + (size_t)i * KDIM))[lane] = o;
}

// ---------------- global add pool (one wave per node) ---------------------
__global__ __launch_bounds__(256)
void k_pool(const float* __restrict__ h, const int* __restrict__ batch,
            float* __restrict__ g, int n) {
    int i    = (blockIdx.x * blockDim.x + threadIdx.x) >> 5;
    int lane = threadIdx.x & 31;
    if (i >= n) return;
    int gi = batch[i];
    const float4 v = ((const float4*)(h + (size_t)i * KDIM))[lane];
    float* out = g + (size_t)gi * KDIM + lane * 4;
    atomicAdd(out + 0, v.x);
    atomicAdd(out + 1, v.y);
    atomicAdd(out + 2, v.z);
    atomicAdd(out + 3, v.w);
}

static inline size_t alignup(size_t x) { return (x + 255) & ~(size_t)255; }

extern "C" void kernel_launch(void* const* d_in, const int* in_sizes, int n_in,
                              void* d_out, int out_size, void* d_ws, size_t ws_size,
                              hipStream_t stream) {
    const float* x    = (const float*)d_in[0];
    const int*   ei   = (const int*)d_in[1];
    const int*   batch= (const int*)d_in[2];
    const float* W1   = (const float*)d_in[3];
    const float* b1   = (const float*)d_in[4];
    const float* W2   = (const float*)d_in[5];
    const float* b2   = (const float*)d_in[6];
    const float* Wfc  = (const float*)d_in[7];
    const float* bfc  = (const float*)d_in[8];

    const int N   = in_sizes[0] / KDIM;      // 100000 nodes
    const int E   = in_sizes[1] / 2;         // 1.6M edges
    const int OUT = in_sizes[8];             // 64
    const int G   = out_size / OUT;          // 1024 graphs
    const int* src = ei;
    const int* dst = ei + E;

    // Workspace carve-up
    char*  w  = (char*)d_ws;
    size_t nb = (size_t)N * KDIM * sizeof(float);
    float* buf0 = (float*)w; w += alignup(nb);     // h / h1 / h2
    float* buf1 = (float*)w; w += alignup(nb);     // t = h @ W
    float* buf2 = (float*)w; w += alignup(nb);     // scatter accumulator
    float* dis  = (float*)w; w += alignup((size_t)N * sizeof(float));
    float* g    = (float*)w; w += alignup((size_t)G * KDIM * sizeof(float));
    (void)ws_size; (void)n_in;

    const int waves_per_blk = 256 / 32;
    dim3 blk256(256), blk128(128);
    dim3 grid_node((N + waves_per_blk - 1) / waves_per_blk);
    dim3 grid_edge((unsigned)(((long long)E * 32 + 255) / 256));
    dim3 grid_gemmN((N + 63) / 64);
    dim3 grid_gemmG((G + 63) / 64);

    // 1) h = x / ||x||
    k_rownorm<<<grid_node, blk256, 0, stream>>>(x, buf0, N);

    // 2) dis = rsqrt(deg + 1)
    hipMemsetAsync(dis, 0, (size_t)N * sizeof(float), stream);
    k_deg<<<dim3((E + 255) / 256), blk256, 0, stream>>>(dst, dis, E);
    k_dis<<<dim3((N + 255) / 256), blk256, 0, stream>>>(dis, N);

    // 3) conv1: t1 = h @ W1 ; aggregate ; gelu(+b1)
    k_gemm_wmma<128, false><<<grid_gemmN, blk128, 0, stream>>>(buf0, W1, nullptr, buf1, N);
    hipMemsetAsync(buf2, 0, nb, stream);
    k_agg_edges<<<grid_edge, blk256, 0, stream>>>(buf1, src, dst, dis, buf2, E);
    k_finalize<<<grid_node, blk256, 0, stream>>>(buf2, buf1, dis, b1, buf0, N);

    // 4) conv2
    k_gemm_wmma<128, false><<<grid_gemmN, blk128, 0, stream>>>(buf0, W2, nullptr, buf1, N);
    hipMemsetAsync(buf2, 0, nb, stream);
    k_agg_edges<<<grid_edge, blk256, 0, stream>>>(buf1, src, dst, dis, buf2, E);
    k_finalize<<<grid_node, blk256, 0, stream>>>(buf2, buf1, dis, b2, buf0, N);

    // 5) global add pool
    hipMemsetAsync(g, 0, (size_t)G * KDIM * sizeof(float), stream);
    k_pool<<<grid_node, blk256, 0, stream>>>(buf0, batch, g, N);

    // 6) out = gelu(g @ Wfc + bfc)
    k_gemm_wmma<64, true><<<grid_gemmG, blk128, 0, stream>>>(g, Wfc, bfc, (float*)d_out, G);
}